// DynamicSparseAttention_4793183502890
// MI455X (gfx1250) — compile-verified
//
#include <hip/hip_runtime.h>

typedef __attribute__((ext_vector_type(2))) float v2f;
typedef __attribute__((ext_vector_type(8))) float v8f;

#define S_LEN   2048
#define DHEAD   64
#define QTILE   16
#define NWAVE   16
#define TPW     8            // column tiles per wave (128 / 16)
#define SCALE   0.125f       // 1/sqrt(64)
#define NEG_INF (-3.0e38f)

// reduce-buffer slot: 16 rows x 64 cols, row stride padded to 65 to dodge bank conflicts
#define RSLOT   1040         // 16*65
// P staging tile: 16x16, row stride 17
#define PSTRIDE 17

__launch_bounds__(512, 1)
__global__ void dyn_attn_kernel(const float* __restrict__ Q,
                                const float* __restrict__ K,
                                const float* __restrict__ V,
                                float* __restrict__ outO,
                                float* __restrict__ outA,
                                float* __restrict__ outT)
{
    __shared__ float redbuf[8 * RSLOT];     // 33.3 KB: P staging (phase 3) then out-reduce tree
    __shared__ float sSum[NWAVE * 16];
    __shared__ float sSq [NWAVE * 16];
    __shared__ float sMax[NWAVE * 16];
    __shared__ float sPs [NWAVE * 16];
    __shared__ float rInvTau[16], rMaxS[16], rInvSum[16];

    const int bid  = blockIdx.x;
    const int bh   = bid >> 7;        // (b*H+h), 128 q-tiles per head
    const int qt   = bid & 127;
    const int q0   = qt * QTILE;
    const int tid  = threadIdx.x;
    const int w    = tid >> 5;        // wave id 0..15
    const int lane = tid & 31;
    const int half = lane >> 4;       // 0: rows 0..7 of C frag, 1: rows 8..15
    const int lm   = lane & 15;
    const int kb   = half * 2;        // A/B fragment K base within 4-chunk

    const size_t hbase = (size_t)bh * (size_t)(S_LEN * DHEAD);
    const float* qp = Q + hbase + (size_t)q0 * DHEAD;
    const float* kp = K + hbase;
    const float* vp = V + hbase;

    // ---- A fragments for the Q tile (16x64 f32, scale folded in) ----
    // A layout (16x4 f32): lane holds M=lm, K = kb+{0,1} within each 4-chunk
    v2f aF[16];
    #pragma unroll
    for (int c = 0; c < 16; ++c) {
        v2f t = *(const v2f*)(qp + lm * DHEAD + c * 4 + kb);
        aF[c].x = t.x * SCALE;
        aF[c].y = t.y * SCALE;
    }

    // ---- Phase 1: scores S = (Q*scale) @ K^T via v_wmma_f32_16x16x4_f32 ----
    // B fragment: lane holds N=lm (key within tile), K rows kb+{0,1}
    v8f sF[TPW];
    #pragma unroll
    for (int t = 0; t < TPW; ++t) {
        const int keyb = (w * TPW + t) * 16;
        v8f acc = {};
        #pragma unroll
        for (int c = 0; c < 16; ++c) {
            v2f bF = *(const v2f*)(kp + (size_t)(keyb + lm) * DHEAD + c * 4 + kb);
            acc = __builtin_amdgcn_wmma_f32_16x16x4_f32(
                      false, aF[c], false, bF, (short)0, acc, false, false);
        }
        sF[t] = acc;
    }

    // ---- Phase 2a: per-row sum / sumsq / max (wave partials) ----
    float ls[8], lq[8], lx[8];
    #pragma unroll
    for (int r = 0; r < 8; ++r) { ls[r] = 0.f; lq[r] = 0.f; lx[r] = NEG_INF; }
    #pragma unroll
    for (int t = 0; t < TPW; ++t)
        #pragma unroll
        for (int r = 0; r < 8; ++r) {
            float v = sF[t][r];
            ls[r] += v; lq[r] += v * v; lx[r] = fmaxf(lx[r], v);
        }
    // butterfly within each 16-lane half (masks < 16 keep halves independent)
    #pragma unroll
    for (int off = 1; off < 16; off <<= 1)
        #pragma unroll
        for (int r = 0; r < 8; ++r) {
            ls[r] += __shfl_xor(ls[r], off, 32);
            lq[r] += __shfl_xor(lq[r], off, 32);
            lx[r]  = fmaxf(lx[r], __shfl_xor(lx[r], off, 32));
        }
    if ((lane & 15) == 0) {
        const int base = w * 16 + half * 8;
        #pragma unroll
        for (int r = 0; r < 8; ++r) {
            sSum[base + r] = ls[r]; sSq[base + r] = lq[r]; sMax[base + r] = lx[r];
        }
    }
    __syncthreads();

    // ---- Phase 2b: cross-wave stats -> var (ddof=1) -> tau ----
    if (tid < 16) {
        float s = 0.f, q2 = 0.f, m = NEG_INF;
        #pragma unroll
        for (int ww = 0; ww < NWAVE; ++ww) {
            s  += sSum[ww * 16 + tid];
            q2 += sSq [ww * 16 + tid];
            m   = fmaxf(m, sMax[ww * 16 + tid]);
        }
        const float mean = s * (1.0f / (float)S_LEN);
        const float var  = (q2 - s * mean) * (1.0f / (float)(S_LEN - 1));
        float tau = 1.0f / (1.0f + var);
        tau = fmaxf(tau, 0.3f);
        rInvTau[tid] = 1.0f / tau;
        rMaxS[tid]   = m;
        outT[(size_t)bh * S_LEN + q0 + tid] = tau;
    }
    __syncthreads();

    // ---- Phase 2c: exp((s - max)/tau), row sums ----
    float it[8], mo[8], ps[8];
    #pragma unroll
    for (int r = 0; r < 8; ++r) {
        const int row = half * 8 + r;
        it[r] = rInvTau[row];
        mo[r] = rMaxS[row] * it[r];
        ps[r] = 0.f;
    }
    #pragma unroll
    for (int t = 0; t < TPW; ++t)
        #pragma unroll
        for (int r = 0; r < 8; ++r) {
            float p = __expf(sF[t][r] * it[r] - mo[r]);
            sF[t][r] = p;
            ps[r]   += p;
        }
    #pragma unroll
    for (int off = 1; off < 16; off <<= 1)
        #pragma unroll
        for (int r = 0; r < 8; ++r) ps[r] += __shfl_xor(ps[r], off, 32);
    if ((lane & 15) == 0) {
        const int base = w * 16 + half * 8;
        #pragma unroll
        for (int r = 0; r < 8; ++r) sPs[base + r] = ps[r];
    }
    __syncthreads();
    if (tid < 16) {
        float s = 0.f;
        #pragma unroll
        for (int ww = 0; ww < NWAVE; ++ww) s += sPs[ww * 16 + tid];
        rInvSum[tid] = 1.0f / s;
    }
    __syncthreads();

    // ---- normalize + non-temporal store of attn (1 GiB stream, keep L2 for K/V) ----
    float nf[8];
    #pragma unroll
    for (int r = 0; r < 8; ++r) nf[r] = rInvSum[half * 8 + r];
    float* ap = outA + (size_t)bh * S_LEN * S_LEN + (size_t)q0 * S_LEN;
    #pragma unroll
    for (int t = 0; t < TPW; ++t) {
        const int colb = (w * TPW + t) * 16 + lm;
        #pragma unroll
        for (int r = 0; r < 8; ++r) {
            float v = sF[t][r] * nf[r];
            sF[t][r] = v;
            __builtin_nontemporal_store(v, ap + (size_t)(half * 8 + r) * S_LEN + colb);
        }
    }

    // ---- Phase 3: out partial = P @ V (wave-local LDS staging, WMMA) ----
    v8f oC[4] = {};
    float* stg = redbuf + w * (16 * PSTRIDE);
    #pragma unroll 1
    for (int t = 0; t < TPW; ++t) {
        const int T = w * TPW + t;
        asm volatile("s_wait_dscnt 0" ::: "memory");   // WAR vs previous iter reads
        #pragma unroll
        for (int r = 0; r < 8; ++r)
            stg[(half * 8 + r) * PSTRIDE + lm] = sF[t][r];
        asm volatile("s_wait_dscnt 0" ::: "memory");   // RAW before A-frag reads
        #pragma unroll
        for (int kc = 0; kc < 4; ++kc) {
            v2f a;
            a.x = stg[lm * PSTRIDE + kc * 4 + kb];
            a.y = stg[lm * PSTRIDE + kc * 4 + kb + 1];
            const int key = T * 16 + kc * 4 + kb;
            #pragma unroll
            for (int nt = 0; nt < 4; ++nt) {
                v2f b;
                b.x = vp[(size_t)key * DHEAD + nt * 16 + lm];
                b.y = vp[(size_t)(key + 1) * DHEAD + nt * 16 + lm];
                oC[nt] = __builtin_amdgcn_wmma_f32_16x16x4_f32(
                             false, a, false, b, (short)0, oC[nt], false, false);
            }
        }
    }

    // ---- Phase 4: deterministic cross-wave reduce of 16x64 out tile ----
    __syncthreads();   // staging region reused as reduce buffer
    #pragma unroll
    for (int stride = 8; stride >= 1; stride >>= 1) {
        if (w >= stride && w < 2 * stride) {
            float* b = redbuf + (w - stride) * RSLOT;
            #pragma unroll
            for (int nt = 0; nt < 4; ++nt)
                #pragma unroll
                for (int r = 0; r < 8; ++r)
                    b[(half * 8 + r) * 65 + nt * 16 + lm] = oC[nt][r];
        }
        __syncthreads();
        if (w < stride) {
            const float* b = redbuf + w * RSLOT;
            #pragma unroll
            for (int nt = 0; nt < 4; ++nt)
                #pragma unroll
                for (int r = 0; r < 8; ++r)
                    oC[nt][r] += b[(half * 8 + r) * 65 + nt * 16 + lm];
        }
        __syncthreads();
    }
    if (w == 0) {
        float* op = outO + hbase + (size_t)q0 * DHEAD;
        #pragma unroll
        for (int nt = 0; nt < 4; ++nt)
            #pragma unroll
            for (int r = 0; r < 8; ++r)
                op[(size_t)(half * 8 + r) * DHEAD + nt * 16 + lm] = oC[nt][r];
    }
}

extern "C" void kernel_launch(void* const* d_in, const int* in_sizes, int n_in,
                              void* d_out, int out_size, void* d_ws, size_t ws_size,
                              hipStream_t stream) {
    const float* q = (const float*)d_in[0];
    const float* k = (const float*)d_in[1];
    const float* v = (const float*)d_in[2];

    float* out  = (float*)d_out;
    float* outO = out;                                          // [4,16,2048,64]
    float* outA = outO + (size_t)4 * 16 * 2048 * 64;            // [4,16,2048,2048]
    float* outT = outA + (size_t)4 * 16 * 2048 * 2048;          // [4,16,2048,1]

    const int nBlocks = 4 * 16 * (S_LEN / QTILE);               // 8192
    dyn_attn_kernel<<<nBlocks, 512, 0, stream>>>(q, k, v, outO, outA, outT);
}